// BalancedCELoss_64561948393810
// MI455X (gfx1250) — compile-verified
//
#include <hip/hip_runtime.h>

typedef __attribute__((ext_vector_type(2))) float v2f;
typedef __attribute__((ext_vector_type(8))) float v8f;

#define NCLS  16
#define SB    (64 * 128 * 128)   // spatial voxels per sample
#define BATCH 2
#define KANN  4
#define EPSV  1e-6f

// Input is pre-clamped to [1e-6, 1-1e-6] -- always normal, so use the raw
// hardware v_log_f32 (log2) and scale by ln(2). Avoids the libm denorm
// pre-scale/post-fix sequence (saves ~4 VALU per log, 9 logs/lane/tile).
__device__ __forceinline__ float clamped_log(float p) {
    p = fminf(fmaxf(p, EPSV), 1.0f - EPSV);   // -> single v_med3_num_f32
    return __builtin_amdgcn_logf(p) * 0.6931471805599453f;
}

__global__ void init_ws_kernel(double* __restrict__ acc, int* __restrict__ flags) {
    int t = threadIdx.x;
    if (t < 3) acc[t] = 0.0;
    if (t < BATCH) flags[t] = 0;
}

__global__ __launch_bounds__(256) void fused_loss_kernel(
    const float* __restrict__ probs, const int* __restrict__ target,
    const int* __restrict__ ann, double* __restrict__ acc, int* __restrict__ flags)
{
    const int b    = blockIdx.y;
    const int tid  = threadIdx.x;
    const int lane = tid & 31;
    const int wave = tid >> 5;
    const int h    = lane >> 4;   // half-wave: which 2-of-4 K slice we own
    const int n    = lane & 15;   // voxel column within 16-voxel tile

    // --- unannotated-class bitmask for this sample (class 0 always unannotated) ---
    unsigned maskbits = 0;
    #pragma unroll
    for (int c = 0; c < NCLS; ++c) {
        bool annot = false;
        #pragma unroll
        for (int k = 0; k < KANN; ++k) {
            int a = ann[b * KANN + k];
            annot = annot || ((a == c) && (a > 0));
        }
        if (!annot) maskbits |= (1u << c);
    }

    // --- A matrix (mask replicated down rows), constant across the whole loop ---
    // ISA 32-bit A 16x4 layout: lanes 0-15 hold K=0,1 ; lanes 16-31 hold K=2,3
    v2f avec[4];
    #pragma unroll
    for (int s = 0; s < 4; ++s) {
        const int c0 = 4 * s + 2 * h;
        avec[s].x = ((maskbits >> c0)       & 1u) ? 1.0f : 0.0f;
        avec[s].y = ((maskbits >> (c0 + 1)) & 1u) ? 1.0f : 0.0f;
    }

    const float* pb = probs  + (size_t)b * NCLS * SB;
    const int*   tb = target + (size_t)b * SB;

    const int wgid   = blockIdx.x * (256 >> 5) + wave;
    const int nwaves = gridDim.x * (256 >> 5);

    float ce_acc  = 0.0f;
    float ent_acc = 0.0f;
    int   anyfg   = 0;

    for (int base = wgid * 16; base < SB; base += nwaves * 16) {
        const int vox = base + n;
        const int t   = tb[vox];

        float entp = 0.0f;   // partial entropy over this lane's 8 classes
        float pf   = 0.0f;   // p[target] if this lane owns it
        v2f bv[4];
        #pragma unroll
        for (int s = 0; s < 4; ++s) {
            const int   c0 = 4 * s + 2 * h;
            const float p0 = pb[(size_t)c0       * SB + vox];
            const float p1 = pb[(size_t)(c0 + 1) * SB + vox];
            entp += p0 * clamped_log(p0) + p1 * clamped_log(p1);
            pf = (t == c0)     ? p0 : pf;
            pf = (t == c0 + 1) ? p1 : pf;
            bv[s].x = p0;
            bv[s].y = p1;
        }

        // s0[n] = sum over unannotated classes of p[c][n], via 4x WMMA f32 16x16x4:
        // D = A(mask,16x4) * B(probs,4x16) + C ; every row of D == s0, so element 0
        // of the accumulator on every lane is s0 for voxel (lane & 15).
        v8f c8 = {};
        c8 = __builtin_amdgcn_wmma_f32_16x16x4_f32(false, avec[0], false, bv[0], (short)0, c8, false, false);
        c8 = __builtin_amdgcn_wmma_f32_16x16x4_f32(false, avec[1], false, bv[1], (short)0, c8, false, false);
        c8 = __builtin_amdgcn_wmma_f32_16x16x4_f32(false, avec[2], false, bv[2], (short)0, c8, false, false);
        c8 = __builtin_amdgcn_wmma_f32_16x16x4_f32(false, avec[3], false, bv[3], (short)0, c8, false, false);
        const float s0 = c8[0];

        // combine the two half-wave partials (each lane then holds full values)
        const float ent_v = entp + __shfl_xor(entp, 16, 32);
        const float pf_v  = pf   + __shfl_xor(pf,   16, 32);

        const float p = (t > 0) ? pf_v : s0;
        const float u = 1.0f - p;
        ce_acc  -= u * u * clamped_log(p);   // focal CE voxel term (GAMMA=2)
        ent_acc += ent_v;
        anyfg   |= (t > 0) ? 1 : 0;
    }

    // --- wave reduction (values are duplicated across half-waves -> x0.5 later) ---
    #pragma unroll
    for (int off = 16; off > 0; off >>= 1) {
        ce_acc  += __shfl_down(ce_acc,  off, 32);
        ent_acc += __shfl_down(ent_acc, off, 32);
    }

    __shared__ double sCE[8];
    __shared__ double sENT[8];
    __shared__ int    sFG;
    if (tid == 0) sFG = 0;
    __syncthreads();
    if (anyfg) atomicOr(&sFG, 1);
    if (lane == 0) {
        sCE[wave]  = (double)ce_acc;
        sENT[wave] = (double)ent_acc;
    }
    __syncthreads();
    if (tid == 0) {
        double ce = 0.0, en = 0.0;
        #pragma unroll
        for (int w = 0; w < 8; ++w) { ce += sCE[w]; en += sENT[w]; }
        atomicAdd(&acc[0], ce);
        atomicAdd(&acc[1 + b], en);
        if (sFG) atomicOr(&flags[b], 1);
    }
}

__global__ void finalize_kernel(const double* __restrict__ acc,
                                const int* __restrict__ flags,
                                float* __restrict__ out)
{
    if (threadIdx.x == 0 && blockIdx.x == 0) {
        const double half = 0.5;  // undo half-wave duplication
        const double ce   = half * acc[0] / ((double)BATCH * (double)SB);
        const double ent0 = half * acc[1] / (double)SB;
        const double ent1 = half * acc[2] / (double)SB;
        const double m0   = flags[0] ? 1.0 : 3.0;  // all-background -> 3x
        const double m1   = flags[1] ? 1.0 : 3.0;
        const double reg  = -(m0 * ent0 + m1 * ent1) / (double)BATCH;
        out[0] = (float)ce;
        out[1] = (float)reg;
    }
}

extern "C" void kernel_launch(void* const* d_in, const int* in_sizes, int n_in,
                              void* d_out, int out_size, void* d_ws, size_t ws_size,
                              hipStream_t stream) {
    const float* probs  = (const float*)d_in[0];
    const int*   target = (const int*)d_in[1];
    const int*   ann    = (const int*)d_in[2];
    float*  out   = (float*)d_out;
    double* acc   = (double*)d_ws;
    int*    flags = (int*)((char*)d_ws + 3 * sizeof(double));

    hipLaunchKernelGGL(init_ws_kernel, dim3(1), dim3(32), 0, stream, acc, flags);

    // 1024 blocks x 2 samples, 256 threads (8 wave32) per block:
    // 8192 waves/sample, 65536 16-voxel tiles/sample -> exactly 8 uniform iters.
    dim3 grid(1024, BATCH, 1);
    hipLaunchKernelGGL(fused_loss_kernel, grid, dim3(256), 0, stream,
                       probs, target, ann, acc, flags);

    hipLaunchKernelGGL(finalize_kernel, dim3(1), dim3(32), 0, stream, acc, flags, out);
}